// SpikeFP32GELUExact_11450382811507
// MI455X (gfx1250) — compile-verified
//
#include <hip/hip_runtime.h>
#include <math.h>

// ---------------------------------------------------------------------------
// SpikeFP32GELUExact for gfx1250 (MI455X)
//   [N, 32] float pulse bits -> decode u32 -> f32 -> exact fp64 tanh-GELU
//   -> f32 -> encode [N, 32] float pulse bits.
// Memory bound: 1 GiB traffic, ~46 us floor @ 23.3 TB/s.
// Data path: coalesced global -> LDS via CDNA5 async-to-LDS, padded LDS
// layout (stride 33 dwords) for conflict-free per-lane strided reads.
// ---------------------------------------------------------------------------

#define VALS_PER_BLOCK 256
#define LDS_STRIDE     33   // 32 + 1 pad; 33*dt % 64 != 0 for dt in 1..31

#if defined(__AMDGCN__) && __has_builtin(__builtin_amdgcn_global_load_async_to_lds_b32)
#define USE_ASYNC_LDS 1
#else
#define USE_ASYNC_LDS 0
#endif

#if USE_ASYNC_LDS
#define AS_GLOBAL __attribute__((address_space(1)))
#define AS_LOCAL  __attribute__((address_space(3)))
#endif

typedef float vfloat4 __attribute__((ext_vector_type(4)));

__global__ __launch_bounds__(VALS_PER_BLOCK) void
gelu_pulse_kernel(const float* __restrict__ in, float* __restrict__ out, int n_vals) {
    __shared__ float tile[VALS_PER_BLOCK * LDS_STRIDE];

    const int t = threadIdx.x;
    const int blockVal0 = blockIdx.x * VALS_PER_BLOCK;

    int nv = n_vals - blockVal0;
    if (nv <= 0) return;                 // uniform per block: no barrier hazard
    if (nv > VALS_PER_BLOCK) nv = VALS_PER_BLOCK;

    const float* gsrc = in + (size_t)blockVal0 * 32;
    const int ndw = nv * 32;             // dwords in this tile

    // ---- Stage 1: coalesced global -> LDS (padded), async on CDNA5 ----
#pragma unroll
    for (int k = 0; k < 32; ++k) {
        int i = k * VALS_PER_BLOCK + t;  // linear dword: perfectly coalesced
        if (i < ndw) {
            int v = i >> 5;              // value within tile
            int j = i & 31;              // bit index within value
#if USE_ASYNC_LDS
            __builtin_amdgcn_global_load_async_to_lds_b32(
                (AS_GLOBAL int*)(gsrc + i),
                (AS_LOCAL  int*)&tile[v * LDS_STRIDE + j],
                /*imm offset*/ 0, /*cpol*/ 0);
#else
            tile[v * LDS_STRIDE + j] = gsrc[i];
#endif
        }
    }

#if USE_ASYNC_LDS
#if __has_builtin(__builtin_amdgcn_s_wait_asynccnt)
    __builtin_amdgcn_s_wait_asynccnt(0);
#else
    asm volatile("s_wait_asynccnt 0" ::: "memory");
#endif
#endif
    __syncthreads();

    // ---- Stage 2: one value per lane, conflict-free strided LDS reads ----
    if (t < nv) {
        const float* lw = &tile[t * LDS_STRIDE];

        // Decode: pulse j is bit (31-j); pulse is exactly 0x3F800000 or 0x0,
        // so bit = (as_uint >> 23) & 1 (no float compare needed).
        unsigned u = 0u;
#pragma unroll
        for (int j = 0; j < 32; ++j) {
            u = (u << 1) | ((__float_as_uint(lw[j]) >> 23) & 1u);
        }

        // Exact fp64 tanh-GELU, mirroring the reference expression order.
        double x     = (double)__uint_as_float(u);
        double x3    = (x * x) * x;
        double inner = x + 0.044715 * x3;
        double two_z = 2.0 * (0.7978845608028654 * inner);
        double e     = exp(two_z);
        double th    = (e - 1.0) / (e + 1.0);
        double y     = 0.5 * (x * (1.0 + th));
        unsigned r   = __float_as_uint((float)y);

        // Encode: out[val*32 + j] = (r >> (31-j)) & 1 as float.
        // Wave writes a contiguous 4KB region; NT stores keep L2 clean.
        vfloat4* dst = (vfloat4*)(out + (size_t)(blockVal0 + t) * 32);
#pragma unroll
        for (int k = 0; k < 8; ++k) {
            vfloat4 f;
            f.x = (float)((r >> (31 - (4 * k + 0))) & 1u);
            f.y = (float)((r >> (31 - (4 * k + 1))) & 1u);
            f.z = (float)((r >> (31 - (4 * k + 2))) & 1u);
            f.w = (float)((r >> (31 - (4 * k + 3))) & 1u);
            __builtin_nontemporal_store(f, &dst[k]);
        }
    }
}

extern "C" void kernel_launch(void* const* d_in, const int* in_sizes, int n_in,
                              void* d_out, int out_size, void* d_ws, size_t ws_size,
                              hipStream_t stream) {
    const float* x  = (const float*)d_in[0];
    float*      yo  = (float*)d_out;
    const int n_vals = in_sizes[0] / 32;                 // 1024*4096 = 4,194,304
    const int blocks = (n_vals + VALS_PER_BLOCK - 1) / VALS_PER_BLOCK;
    gelu_pulse_kernel<<<blocks, VALS_PER_BLOCK, 0, stream>>>(x, yo, n_vals);
}